// GroupNLMSMemory_9234179687032
// MI455X (gfx1250) — compile-verified
//
#include <hip/hip_runtime.h>
#include <math.h>

#define B_SZ 512
#define M_SZ 100000
#define D_SZ 256
#define EPS_C 1e-8f
#define TEMP_C 10.0f
#define KSPLIT 16

typedef float v2f __attribute__((ext_vector_type(2)));
typedef float v8f __attribute__((ext_vector_type(8)));

// ---------------------------------------------------------------------------
// CDNA5 async global->LDS copy (ASYNCcnt-tracked), with safe fallback
// Builtin signature (from hipcc diagnostic): b128 variant takes pointers to
// 16-byte int vectors: (AS1 v4i*, AS3 v4i*, imm offset, imm cpol).
// ---------------------------------------------------------------------------
#if defined(__HIP_DEVICE_COMPILE__) && defined(__has_builtin)
# if __has_builtin(__builtin_amdgcn_global_load_async_to_lds_b128)
#  define HAVE_ASYNC_LDS 1
# endif
#endif
#ifndef HAVE_ASYNC_LDS
# define HAVE_ASYNC_LDS 0
#endif

typedef int v4i_vs __attribute__((vector_size(16)));
typedef __attribute__((address_space(1))) v4i_vs Gv4i;
typedef __attribute__((address_space(3))) v4i_vs Lv4i;

__device__ __forceinline__ void copy16_g2l(const float* g, float* l) {
#if HAVE_ASYNC_LDS
    __builtin_amdgcn_global_load_async_to_lds_b128((Gv4i*)g, (Lv4i*)l, 0, 0);
#else
    float4 v = *(const float4*)g;
    *(float4*)l = v;
#endif
}

__device__ __forceinline__ void wait_async0() {
#if HAVE_ASYNC_LDS
    asm volatile("s_wait_asynccnt 0x0" ::: "memory");
#endif
}

// ---------------------------------------------------------------------------
// 1) Row norms: one wave32 per row (keys rows first, then x rows)
// ---------------------------------------------------------------------------
__global__ void norms_kernel(const float* __restrict__ x,
                             const float* __restrict__ keys,
                             float* __restrict__ kn,
                             float* __restrict__ xn) {
    int wave = (blockIdx.x * blockDim.x + threadIdx.x) >> 5;
    int lane = threadIdx.x & 31;
    const float* row;
    if (wave < M_SZ)               row = keys + (size_t)wave * D_SZ;
    else if (wave < M_SZ + B_SZ)   row = x + (size_t)(wave - M_SZ) * D_SZ;
    else return;

    const float4* r4 = (const float4*)row;   // 256 floats = 64 float4
    float4 a = r4[lane];
    float4 b = r4[lane + 32];
    float s = a.x*a.x + a.y*a.y + a.z*a.z + a.w*a.w
            + b.x*b.x + b.y*b.y + b.z*b.z + b.w*b.w;
    #pragma unroll
    for (int off = 16; off > 0; off >>= 1) s += __shfl_xor(s, off, 32);
    if (lane == 0) {
        float n = sqrtf(s);
        if (wave < M_SZ) kn[wave] = n;
        else             xn[wave - M_SZ] = n;
    }
}

// ---------------------------------------------------------------------------
// 2) GEMM1: logits[b][m] = dot(x[b], keys[m]) / max(xn*kn, eps) * TEMP
//    grid = (16 row-groups [fastest, share keys chunk in L2], 782 m-blocks)
//    block = 256 thr (8 waves) -> 32 (b) x 128 (m) tile, fp32 WMMA 16x16x4
//    keys chunk double-buffered via async global->LDS copies
// ---------------------------------------------------------------------------
#define KCH 16
__global__ __launch_bounds__(256) void gemm1_kernel(
        const float* __restrict__ x, const float* __restrict__ keys,
        const float* __restrict__ xn, const float* __restrict__ kn,
        float* __restrict__ logits) {
    __shared__ float xs[32][D_SZ + 4];     // stride 260 fl = 1040B (16B aligned)
    __shared__ float ks[2][128][KCH + 4];  // stride 20 fl  = 80B   (16B aligned)

    const int tid  = threadIdx.x;
    const int lane = tid & 31, wave = tid >> 5;
    const int lo = lane & 15, hi = lane >> 4;
    const int wrow = wave & 1;             // 2 row groups of 16
    const int wcol = wave >> 1;            // 4 col groups of 32
    const int b0 = blockIdx.x * 32;
    const int m0 = blockIdx.y * 128;

    // edge m-block: pre-zero both key buffers so OOB rows contribute zeros
    if (m0 + 128 > M_SZ) {
        float* p = (float*)ks;
        for (int i = tid; i < 2 * 128 * (KCH + 4); i += 256) p[i] = 0.f;
        __syncthreads();
    }

    auto stage_ks = [&](int buf, int kb) {
        #pragma unroll
        for (int p = 0; p < 2; ++p) {       // 128 rows x 16 K = 512 float4
            int ml  = p * 64 + (tid >> 2);
            int kk4 = (tid & 3) * 4;
            int gm  = m0 + ml;
            if (gm < M_SZ)
                copy16_g2l(keys + (size_t)gm * D_SZ + kb + kk4, &ks[buf][ml][kk4]);
        }
    };

    // stage x tile 32x256 once (async; exactly in range)
    #pragma unroll
    for (int p = 0; p < 8; ++p) {
        int bl = p * 4 + (tid >> 6);
        int k  = (tid & 63) * 4;
        copy16_g2l(x + (size_t)(b0 + bl) * D_SZ + k, &xs[bl][k]);
    }
    stage_ks(0, 0);

    v8f acc0 = {}; v8f acc1 = {};
    #pragma unroll 1
    for (int c = 0; c < D_SZ / KCH; ++c) {
        wait_async0();          // chunk issued last iteration has landed
        __syncthreads();        // visible to all waves; prev compute done
        if (c + 1 < D_SZ / KCH) stage_ks((c + 1) & 1, (c + 1) * KCH);
        const int cur = c & 1;
        const int kb  = c * KCH;
        #pragma unroll
        for (int kk = 0; kk < KCH; kk += 4) {
            v2f a;
            int ar = wrow * 16 + lo;
            a.x = xs[ar][kb + kk + 2 * hi];
            a.y = xs[ar][kb + kk + 2 * hi + 1];
            v2f bf0, bf1;
            int c0 = wcol * 32;
            bf0.x = ks[cur][c0 + lo][kk + 2 * hi];
            bf0.y = ks[cur][c0 + lo][kk + 2 * hi + 1];
            bf1.x = ks[cur][c0 + 16 + lo][kk + 2 * hi];
            bf1.y = ks[cur][c0 + 16 + lo][kk + 2 * hi + 1];
            acc0 = __builtin_amdgcn_wmma_f32_16x16x4_f32(false, a, false, bf0,
                                                         (short)0, acc0, false, false);
            acc1 = __builtin_amdgcn_wmma_f32_16x16x4_f32(false, a, false, bf1,
                                                         (short)0, acc1, false, false);
        }
    }

    // epilogue: C layout -> VGPR j, lane L: row = j + 8*(L/16), col = L%16
    const int gb_base = b0 + wrow * 16;
    #pragma unroll
    for (int j = 0; j < 8; ++j) {
        int gb = gb_base + j + 8 * hi;
        float xv = xn[gb];
        int gm0 = m0 + wcol * 32 + lo;
        if (gm0 < M_SZ) {
            float den = fmaxf(xv * kn[gm0], EPS_C);
            logits[(size_t)gb * M_SZ + gm0] = acc0[j] * (TEMP_C / den);
        }
        int gm1 = gm0 + 16;
        if (gm1 < M_SZ) {
            float den = fmaxf(xv * kn[gm1], EPS_C);
            logits[(size_t)gb * M_SZ + gm1] = acc1[j] * (TEMP_C / den);
        }
    }
}

// ---------------------------------------------------------------------------
// 3) Softmax row stats: single online pass (max + sum) per row
// ---------------------------------------------------------------------------
__global__ void softmax_stats_kernel(const float* __restrict__ logits,
                                     float* __restrict__ rowmax,
                                     float* __restrict__ rowinv) {
    __shared__ float sm[256], ss[256];
    const int b = blockIdx.x, t = threadIdx.x;
    const float4* row4 = (const float4*)(logits + (size_t)b * M_SZ);

    float m = -3.402823466e38f, s = 0.f;
    for (int i = t; i < M_SZ / 4; i += 256) {
        float4 v = row4[i];
        float m4 = fmaxf(fmaxf(v.x, v.y), fmaxf(v.z, v.w));
        if (m4 > m) { s *= expf(m - m4); m = m4; }
        s += expf(v.x - m) + expf(v.y - m) + expf(v.z - m) + expf(v.w - m);
    }
    sm[t] = m; ss[t] = s; __syncthreads();
    for (int st = 128; st > 0; st >>= 1) {
        if (t < st) {
            float ma = sm[t], mb = sm[t + st];
            float M  = fmaxf(ma, mb);
            ss[t] = ss[t] * expf(ma - M) + ss[t + st] * expf(mb - M);
            sm[t] = M;
        }
        __syncthreads();
    }
    if (t == 0) { rowmax[b] = sm[0]; rowinv[b] = 1.0f / ss[0]; }
}

// ---------------------------------------------------------------------------
// 4) Normalize attn in place (float4; M % 4 == 0 so vectors never cross rows)
// ---------------------------------------------------------------------------
__global__ void softmax_norm_kernel(float* __restrict__ attn,
                                    const float* __restrict__ rowmax,
                                    const float* __restrict__ rowinv) {
    size_t idx = ((size_t)blockIdx.x * blockDim.x + threadIdx.x) * 4;
    if (idx >= (size_t)B_SZ * M_SZ) return;
    int b = (int)(idx / M_SZ);
    float mx = rowmax[b], ri = rowinv[b];
    float4 v = *(float4*)(attn + idx);
    v.x = expf(v.x - mx) * ri;
    v.y = expf(v.y - mx) * ri;
    v.z = expf(v.z - mx) * ri;
    v.w = expf(v.w - mx) * ri;
    *(float4*)(attn + idx) = v;
}

__global__ void zero_kernel(float* __restrict__ p, int n) {
    int i = blockIdx.x * blockDim.x + threadIdx.x;
    if (i < n) p[i] = 0.f;
}

// ---------------------------------------------------------------------------
// 5) GEMM2: retrieved[b][d] = sum_m attn[b][m] * V[m][d], split-K over M
//    grid = (32 b-tiles [fastest, share vals chunk in L2], 16 k-splits)
//    block = 256 thr (8 waves) -> 16 (b) x 256 (d=full) tile; attn read once
//    vals + attn tiles double-buffered via async global->LDS copies
// ---------------------------------------------------------------------------
#define MCH2 16
__global__ __launch_bounds__(256) void gemm2_kernel(
        const float* __restrict__ attn, const float* __restrict__ vals,
        float* __restrict__ outr) {
    __shared__ float vs[2][MCH2][D_SZ + 4];   // 2 x 16 x 260 fl (1040B rows)
    __shared__ float as_[2][16][MCH2 + 4];    // 2 x 16 x 20 fl  (80B rows)

    const int tid  = threadIdx.x;
    const int lane = tid & 31, wave = tid >> 5;
    const int lo = lane & 15, hi = lane >> 4;
    const int b0 = blockIdx.x * 16;
    const int mstart = (M_SZ / KSPLIT) * blockIdx.y;
    const int mend   = mstart + (M_SZ / KSPLIT);

    auto stage_vs = [&](int buf, int m) {
        #pragma unroll
        for (int p = 0; p < 4; ++p) {         // 16 rows x 256 d = 1024 float4
            int k = p * 4 + (tid >> 6);
            int c = (tid & 63) * 4;
            int gm = m + k;
            if (gm < mend)                     // tail: stale rows x zero attn = 0
                copy16_g2l(vals + (size_t)gm * D_SZ + c, &vs[buf][k][c]);
        }
    };
    auto stage_as = [&](int buf, int m) {
        if (tid < 64) {                        // 16 rows x 16 m = 64 float4
            int r = tid >> 2, k = (tid & 3) * 4;
            int gm = m + k;
            const float* src = attn + (size_t)(b0 + r) * M_SZ + gm;
            if (gm + 3 < mend) {
                copy16_g2l(src, &as_[buf][r][k]);
            } else {
                #pragma unroll
                for (int q = 0; q < 4; ++q)
                    as_[buf][r][k + q] = (gm + q < mend) ? src[q] : 0.f;
            }
        }
    };

    v8f acc0 = {}, acc1 = {};
    const int nch = (mend - mstart + MCH2 - 1) / MCH2;
    stage_vs(0, mstart);
    stage_as(0, mstart);
    #pragma unroll 1
    for (int c = 0; c < nch; ++c) {
        wait_async0();
        __syncthreads();
        if (c + 1 < nch) {
            stage_vs((c + 1) & 1, mstart + (c + 1) * MCH2);
            stage_as((c + 1) & 1, mstart + (c + 1) * MCH2);
        }
        const int cur = c & 1;
        const int c0  = wave * 32;
        #pragma unroll
        for (int kk = 0; kk < MCH2; kk += 4) {
            v2f a;
            a.x = as_[cur][lo][kk + 2 * hi];
            a.y = as_[cur][lo][kk + 2 * hi + 1];
            v2f bf0, bf1;
            bf0.x = vs[cur][kk + 2 * hi][c0 + lo];
            bf0.y = vs[cur][kk + 2 * hi + 1][c0 + lo];
            bf1.x = vs[cur][kk + 2 * hi][c0 + 16 + lo];
            bf1.y = vs[cur][kk + 2 * hi + 1][c0 + 16 + lo];
            acc0 = __builtin_amdgcn_wmma_f32_16x16x4_f32(false, a, false, bf0,
                                                         (short)0, acc0, false, false);
            acc1 = __builtin_amdgcn_wmma_f32_16x16x4_f32(false, a, false, bf1,
                                                         (short)0, acc1, false, false);
        }
    }

    #pragma unroll
    for (int j = 0; j < 8; ++j) {
        int gb = b0 + j + 8 * hi;
        atomicAdd(outr + (size_t)gb * D_SZ + wave * 32 + lo,      acc0[j]);
        atomicAdd(outr + (size_t)gb * D_SZ + wave * 32 + 16 + lo, acc1[j]);
    }
}

// ---------------------------------------------------------------------------
extern "C" void kernel_launch(void* const* d_in, const int* in_sizes, int n_in,
                              void* d_out, int out_size, void* d_ws, size_t ws_size,
                              hipStream_t stream) {
    const float* x    = (const float*)d_in[0];
    const float* keys = (const float*)d_in[1];
    const float* vals = (const float*)d_in[2];
    float* out       = (float*)d_out;
    float* retrieved = out;                              // [B, D]
    float* attn      = out + (size_t)B_SZ * D_SZ;        // [B, M]

    float* ws     = (float*)d_ws;
    float* kn     = ws;               // M floats (padded region 100352)
    float* xn     = ws + 100352;      // B floats
    float* rowmax = xn + 512;         // B floats
    float* rowinv = rowmax + 512;     // B floats

    // 1) norms
    int nwaves = M_SZ + B_SZ;
    norms_kernel<<<(nwaves + 7) / 8, 256, 0, stream>>>(x, keys, kn, xn);

    // 2) logits via fp32 WMMA (row-groups fastest -> keys chunks L2-resident)
    gemm1_kernel<<<dim3(B_SZ / 32, (M_SZ + 127) / 128), 256, 0, stream>>>(
        x, keys, xn, kn, attn);

    // 3) one-pass softmax stats + 4) normalize
    softmax_stats_kernel<<<B_SZ, 256, 0, stream>>>(attn, rowmax, rowinv);
    {
        size_t nvec = (size_t)B_SZ * M_SZ / 4;
        softmax_norm_kernel<<<(unsigned)((nvec + 255) / 256), 256, 0, stream>>>(
            attn, rowmax, rowinv);
    }

    // 5) retrieved via fp32 WMMA with split-K atomic reduction
    zero_kernel<<<(B_SZ * D_SZ + 255) / 256, 256, 0, stream>>>(retrieved, B_SZ * D_SZ);
    gemm2_kernel<<<dim3(B_SZ / 16, KSPLIT), 256, 0, stream>>>(attn, vals, retrieved);
}